// GraphKANEncoder_25812753449327
// MI455X (gfx1250) — compile-verified
//
#include <hip/hip_runtime.h>
#include <hip/hip_bf16.h>

// GraphKAN encoder for MI455X (gfx1250, wave32).
// Memory-bound workload: sparse props dominate traffic; dense GEMMs run on
// V_WMMA_F32_16X16X4_F32 (fp32 WMMA) with LDS-staged A strips.

typedef __attribute__((ext_vector_type(2))) float v2f;
typedef __attribute__((ext_vector_type(8))) float v8f;

#define D_FEAT 128
#define A_FEAT 16
#define G_POOL 64
#define LN_EPS 1e-5f

// ---------------------------------------------------------------- utilities

__global__ __launch_bounds__(256) void copy_f32(const float* __restrict__ src,
                                                float* __restrict__ dst, int n) {
    int i = blockIdx.x * 256 + threadIdx.x;
    if (i < n) dst[i] = src[i];
}

__global__ __launch_bounds__(256) void fill_f32(float* __restrict__ p, float v, int n) {
    int i = blockIdx.x * 256 + threadIdx.x;
    if (i < n) p[i] = v;
}

// ------------------------------------------------- edge weights / laplacian

// w[e] = sigmoid(edge_attr[e] . ew + eb);  deg[dst[e]] += w[e]  (deg pre-filled 1.0)
__global__ __launch_bounds__(256) void edge_w_deg(const float* __restrict__ ea,
                                                  const float* __restrict__ ew,
                                                  const float* __restrict__ eb,
                                                  const int* __restrict__ dst,
                                                  float* __restrict__ w,
                                                  float* __restrict__ deg, int E) {
    int e = blockIdx.x * 256 + threadIdx.x;
    if (e >= E) return;
    const float4* ea4 = (const float4*)(ea) + (size_t)e * 4;
    const float4* ew4 = (const float4*)ew;
    float s = eb[0];
#pragma unroll
    for (int j = 0; j < 4; ++j) {
        float4 a = ea4[j];
        float4 k = ew4[j];
        s += a.x * k.x + a.y * k.y + a.z * k.z + a.w * k.w;
    }
    float wv = 1.0f / (1.0f + __expf(-s));
    w[e] = wv;
    unsafeAtomicAdd(&deg[dst[e]], wv);
}

// dinv = rsqrt(deg); lam_self = -1/deg
__global__ __launch_bounds__(256) void node_deg_terms(const float* __restrict__ deg,
                                                      float* __restrict__ dinv,
                                                      float* __restrict__ lamself, int N) {
    int i = blockIdx.x * 256 + threadIdx.x;
    if (i >= N) return;
    float d = deg[i];
    dinv[i] = rsqrtf(d);
    lamself[i] = -1.0f / d;
}

// lam[e] = -(dinv[src] * w * dinv[dst])
__global__ __launch_bounds__(256) void edge_lam(const float* __restrict__ w,
                                                const float* __restrict__ dinv,
                                                const int* __restrict__ src,
                                                const int* __restrict__ dst,
                                                float* __restrict__ lam, int E) {
    int e = blockIdx.x * 256 + threadIdx.x;
    if (e >= E) return;
    lam[e] = -(dinv[src[e]] * w[e] * dinv[dst[e]]);
}

// -------------------------------------------------------------- propagation

// t = scale * lam_self[n] * h + pc * prev   (elementwise over N*128)
__global__ __launch_bounds__(256) void prop_init(float* __restrict__ t,
                                                 const float* __restrict__ h,
                                                 const float* __restrict__ prev,
                                                 const float* __restrict__ lamself,
                                                 float scale, float pc, int n) {
    int i = blockIdx.x * 256 + threadIdx.x;
    if (i >= n) return;
    t[i] = scale * lamself[i >> 7] * h[i] + pc * prev[i];
}

// t[dst[e]][d] += scale * lam[e] * h[src[e]][d]
__global__ __launch_bounds__(256) void prop_edges(float* __restrict__ t,
                                                  const float* __restrict__ h,
                                                  const float* __restrict__ lam,
                                                  const int* __restrict__ src,
                                                  const int* __restrict__ dst,
                                                  float scale, int E) {
    int idx = blockIdx.x * 256 + threadIdx.x;
    if (idx >= E * D_FEAT) return;
    int e = idx >> 7;
    int d = idx & 127;
    float coef = scale * lam[e];
    unsafeAtomicAdd(&t[(size_t)dst[e] * D_FEAT + d], coef * h[(size_t)src[e] * D_FEAT + d]);
}

// ---------------------------------------------------------------- WMMA GEMM
// C[16 x 128 strip] (+)= A[16 x 128 strip] @ W[128 x 128]   via f32 16x16x4 WMMA.
// 256 threads = 8 waves; wave w owns output column tile [16w, 16w+16).
// Requires N % 16 == 0 (holds: 50000 = 3125*16) so EXEC is all-ones.
__global__ __launch_bounds__(256) void gemm16_wmma(const float* __restrict__ A,
                                                   const float* __restrict__ W,
                                                   const float* __restrict__ bias,
                                                   float* __restrict__ C,
                                                   int accumulate) {
    __shared__ float sA[16 * D_FEAT];  // 8 KB A strip
    const int rowBase = blockIdx.x * 16;
    const int tid = threadIdx.x;

    // stage A strip (2048 floats, coalesced)
    const float* Ab = A + (size_t)rowBase * D_FEAT;
#pragma unroll
    for (int i = 0; i < 8; ++i) sA[tid + 256 * i] = Ab[tid + 256 * i];
    __syncthreads();

    const int wave = tid >> 5;   // column tile
    const int lane = tid & 31;
    const int m = lane & 15;
    const int kh = lane >> 4;    // 0/1 -> which K half of the 4-wide fragment
    const int col = wave * 16 + m;

    v8f c;
    if (accumulate) {
#pragma unroll
        for (int r = 0; r < 8; ++r)
            c[r] = C[(size_t)(rowBase + r + 8 * kh) * D_FEAT + col];
    } else {
        float b = bias[col];
#pragma unroll
        for (int r = 0; r < 8; ++r) c[r] = b;
    }

#pragma unroll 8
    for (int kb = 0; kb < D_FEAT; kb += 4) {
        v2f a, b;
        // A fragment: lane holds A[m][kb+2*kh], A[m][kb+2*kh+1]   (from LDS)
        a.x = sA[m * D_FEAT + kb + 2 * kh];
        a.y = sA[m * D_FEAT + kb + 2 * kh + 1];
        // B fragment: rows of W striped across lanes
        b.x = W[(size_t)(kb + 2 * kh) * D_FEAT + col];
        b.y = W[(size_t)(kb + 2 * kh + 1) * D_FEAT + col];
        c = __builtin_amdgcn_wmma_f32_16x16x4_f32(false, a, false, b,
                                                  (short)0, c, false, false);
    }

#pragma unroll
    for (int r = 0; r < 8; ++r)
        C[(size_t)(rowBase + r + 8 * kh) * D_FEAT + col] = c[r];
}

// --------------------------------------------------------- LayerNorm + ReLU
// one wave per node row; 4 features per lane; wave32 shuffle reduction
__global__ __launch_bounds__(256) void ln_relu_kernel(const float* __restrict__ in,
                                                      const float* __restrict__ g,
                                                      const float* __restrict__ be,
                                                      float* __restrict__ out, int N) {
    int n = blockIdx.x * 8 + (threadIdx.x >> 5);
    int lane = threadIdx.x & 31;
    if (n >= N) return;
    float4 v = ((const float4*)(in + (size_t)n * D_FEAT))[lane];
    float s = v.x + v.y + v.z + v.w;
    float q = v.x * v.x + v.y * v.y + v.z * v.z + v.w * v.w;
#pragma unroll
    for (int off = 16; off >= 1; off >>= 1) {
        s += __shfl_xor(s, off, 32);
        q += __shfl_xor(q, off, 32);
    }
    float mu = s * (1.0f / 128.0f);
    float var = q * (1.0f / 128.0f) - mu * mu;
    float rs = rsqrtf(var + LN_EPS);
    float4 gg = ((const float4*)g)[lane];
    float4 bb = ((const float4*)be)[lane];
    float4 o;
    o.x = fmaxf(0.0f, (v.x - mu) * rs * gg.x + bb.x);
    o.y = fmaxf(0.0f, (v.y - mu) * rs * gg.y + bb.y);
    o.z = fmaxf(0.0f, (v.z - mu) * rs * gg.z + bb.z);
    o.w = fmaxf(0.0f, (v.w - mu) * rs * gg.w + bb.w);
    ((float4*)(out + (size_t)n * D_FEAT))[lane] = o;
}

// ---------------------------------------------------------------- pooling

__global__ __launch_bounds__(256) void pool_acc(const float* __restrict__ h,
                                                const int* __restrict__ batch,
                                                float* __restrict__ pool,
                                                float* __restrict__ cnt, int n) {
    int idx = blockIdx.x * 256 + threadIdx.x;
    if (idx >= n) return;
    int node = idx >> 7;
    int d = idx & 127;
    int gidx = batch[node];
    unsafeAtomicAdd(&pool[(size_t)gidx * D_FEAT + d], h[idx]);
    if (d == 0) unsafeAtomicAdd(&cnt[gidx], 1.0f);
}

__global__ __launch_bounds__(256) void pool_fin(const float* __restrict__ pool,
                                                const float* __restrict__ cnt,
                                                float* __restrict__ out, int n) {
    int i = blockIdx.x * 256 + threadIdx.x;
    if (i >= n) return;
    out[i] = pool[i] / fmaxf(cnt[i >> 7], 1.0f);
}

// ---------------------------------------------------------------- launcher

extern "C" void kernel_launch(void* const* d_in, const int* in_sizes, int n_in,
                              void* d_out, int out_size, void* d_ws, size_t ws_size,
                              hipStream_t stream) {
    const float* x      = (const float*)d_in[0];
    const int*   eidx   = (const int*)d_in[1];
    const float* eattr  = (const float*)d_in[2];
    const int*   batch  = (const int*)d_in[3];

    const int N = in_sizes[0] / D_FEAT;
    const int E = in_sizes[1] / 2;
    const int* srcp = eidx;
    const int* dstp = eidx + E;

    // workspace partition (floats)
    float* ws = (float*)d_ws;
    const size_t ND = (size_t)N * D_FEAT;
    float* H   = ws;               ws += ND;   // current features
    float* T1  = ws;               ws += ND;   // T_{k-1}
    float* T2  = ws;               ws += ND;   // T_k
    float* OUT = ws;               ws += ND;   // layer output accumulator
    float* w_e = ws;               ws += E;    // edge weights
    float* lam = ws;               ws += E;    // -(dinv w dinv)
    float* deg = ws;               ws += N;
    float* dinv = ws;              ws += N;
    float* lamself = ws;           ws += N;
    float* pool = ws;              ws += (size_t)G_POOL * D_FEAT;
    float* cnt = ws;               ws += G_POOL;

    const int ndTot = N * D_FEAT;              // 6.4M, fits int
    const int edTot = E * D_FEAT;              // 102.4M, fits int
    const int bND = (ndTot + 255) / 256;
    const int bN  = (N + 255) / 256;
    const int bE  = (E + 255) / 256;
    const int bED = (edTot + 255) / 256;
    const int bGemm = N / 16;                  // 3125, N % 16 == 0
    const int bLN   = (N + 7) / 8;

    static const int KS_[4] = {1, 2, 2, 3};

    copy_f32<<<bND, 256, 0, stream>>>(x, H, ndTot);

    for (int l = 0; l < 4; ++l) {
        const int K = KS_[l];
        const float* Wl = (const float*)d_in[4 + 6 * l + 0];
        const float* bl = (const float*)d_in[4 + 6 * l + 1];
        const float* ewl = (const float*)d_in[4 + 6 * l + 2];
        const float* ebl = (const float*)d_in[4 + 6 * l + 3];
        const float* gl = (const float*)d_in[4 + 6 * l + 4];
        const float* bel = (const float*)d_in[4 + 6 * l + 5];

        if (K > 1) {
            fill_f32<<<bN, 256, 0, stream>>>(deg, 1.0f, N);  // self-loop weight 1
            edge_w_deg<<<bE, 256, 0, stream>>>(eattr, ewl, ebl, dstp, w_e, deg, E);
            node_deg_terms<<<bN, 256, 0, stream>>>(deg, dinv, lamself, N);
            edge_lam<<<bE, 256, 0, stream>>>(w_e, dinv, srcp, dstp, lam, E);
        }

        // k = 0: OUT = H @ W0 + b
        gemm16_wmma<<<bGemm, 256, 0, stream>>>(H, Wl, bl, OUT, 0);

        if (K > 1) {
            // T1 = L_hat H
            prop_init<<<bND, 256, 0, stream>>>(T1, H, H, lamself, 1.0f, 0.0f, ndTot);
            prop_edges<<<bED, 256, 0, stream>>>(T1, H, lam, srcp, dstp, 1.0f, E);
            gemm16_wmma<<<bGemm, 256, 0, stream>>>(T1, Wl + 1 * D_FEAT * D_FEAT, bl, OUT, 1);

            if (K > 2) {
                // T2 = 2 L_hat T1 - H   (factor 2 and -H folded into init/scatter)
                prop_init<<<bND, 256, 0, stream>>>(T2, T1, H, lamself, 2.0f, -1.0f, ndTot);
                prop_edges<<<bED, 256, 0, stream>>>(T2, T1, lam, srcp, dstp, 2.0f, E);
                gemm16_wmma<<<bGemm, 256, 0, stream>>>(T2, Wl + 2 * D_FEAT * D_FEAT, bl, OUT, 1);
            }
        }

        ln_relu_kernel<<<bLN, 256, 0, stream>>>(OUT, gl, bel, H, N);
    }

    // global mean pool
    fill_f32<<<(G_POOL * D_FEAT + 255) / 256, 256, 0, stream>>>(pool, 0.0f, G_POOL * D_FEAT);
    fill_f32<<<1, 256, 0, stream>>>(cnt, 0.0f, G_POOL);
    pool_acc<<<bND, 256, 0, stream>>>(H, batch, pool, cnt, ndTot);
    pool_fin<<<(G_POOL * D_FEAT + 255) / 256, 256, 0, stream>>>(pool, cnt, (float*)d_out,
                                                               G_POOL * D_FEAT);
}